// GCNConv_11227044512383
// MI455X (gfx1250) — compile-verified
//
#include <hip/hip_runtime.h>
#include <hip/hip_bf16.h>

#define N_NODES 50000
#define N_EDGES 600000
#define N_REL   8
#define D       128          // D_IN == D_OUT
#define N_TILES 3125         // 50000 / 16

typedef __attribute__((ext_vector_type(2))) float v2f;
typedef __attribute__((ext_vector_type(8))) float v8f;

__device__ __forceinline__ v8f wmma4(v2f a, v2f b, v8f c) {
    // D = A(16x4) * B(4x16) + C(16x16), all fp32. 8-arg form:
    // (neg_a, A, neg_b, B, c_mod, C, reuse_a, reuse_b)
    return __builtin_amdgcn_wmma_f32_16x16x4_f32(false, a, false, b, (short)0, c, false, false);
}

// ---------------------------------------------------------------------------
// Kernel 1: per-relation transform. blockIdx.y = rel (0..7 -> xW, 8 -> mid).
// 4 waves/block, one 16-row tile per wave, all 8 column tiles accumulated.
// Weight matrix (128x128) staged in LDS with (k-pair, n) interleaved layout
// so each B fetch is a single 8-byte ds load:  Wlds[(k>>1)*256 + n*2 + (k&1)]
// ---------------------------------------------------------------------------
__global__ __launch_bounds__(128) void k_transform(
    const float* __restrict__ x, const float* __restrict__ Wrel,
    const float* __restrict__ Wloop, const float* __restrict__ brel,
    float* __restrict__ xW, float* __restrict__ mid)
{
    __shared__ float Wlds[D * D];

    const int rel = blockIdx.y;
    const float* W = (rel < N_REL) ? (Wrel + (size_t)rel * D * D) : Wloop;

    // cooperative load + reshuffle into pair layout
    for (int i = threadIdx.x; i < (D * D) / 2; i += blockDim.x) {
        int kp = i >> 7;          // k pair index 0..63
        int n  = i & 127;
        Wlds[i * 2 + 0] = W[(2 * kp + 0) * D + n];
        Wlds[i * 2 + 1] = W[(2 * kp + 1) * D + n];
    }
    __syncthreads();

    const int wave = threadIdx.x >> 5;
    const int lane = threadIdx.x & 31;
    const int tile = blockIdx.x * 4 + wave;
    if (tile >= N_TILES) return;              // wave-uniform guard

    const int m  = lane & 15;                 // row within tile (A) / col (B,C)
    const int hi = lane >> 4;                 // half-wave select
    const int kh = hi * 2;                    // K sub-offset {0,2}
    const int row0 = tile * 16;

    const float* xrow = x + (size_t)(row0 + m) * D;

    v8f acc[8];
    if (rel == N_REL) {
        #pragma unroll
        for (int nt = 0; nt < 8; ++nt) {
            float bv = brel[nt * 16 + m];
            #pragma unroll
            for (int v = 0; v < 8; ++v) acc[nt][v] = bv;
        }
    } else {
        #pragma unroll
        for (int nt = 0; nt < 8; ++nt) acc[nt] = (v8f){};
    }

    for (int kc = 0; kc < 32; ++kc) {
        const int kk = kc * 4 + kh;           // even
        v2f a;
        a[0] = xrow[kk];
        a[1] = xrow[kk + 1];
        const float* bp = Wlds + (kk >> 1) * 256;
        #pragma unroll
        for (int nt = 0; nt < 8; ++nt) {
            v2f b = *(const v2f*)(bp + (nt * 16 + m) * 2);
            acc[nt] = wmma4(a, b, acc[nt]);
        }
    }

    float* dstp = (rel < N_REL) ? (xW + (size_t)rel * N_NODES * D) : mid;
    const int rbase = row0 + hi * 8;
    #pragma unroll
    for (int nt = 0; nt < 8; ++nt) {
        #pragma unroll
        for (int v = 0; v < 8; ++v)
            dstp[(size_t)(rbase + v) * D + nt * 16 + m] = acc[nt][v];
    }
}

// ---------------------------------------------------------------------------
// Kernel 2: edge message gather + atomic scatter-add.
// One wave per edge: 32 lanes x float4 = 128 floats, 4 f32 atomics per lane.
// ---------------------------------------------------------------------------
__global__ __launch_bounds__(256) void k_scatter(
    const float* __restrict__ xW, const int* __restrict__ src,
    const int* __restrict__ dst, const int* __restrict__ etype,
    float* mid)
{
    const int wid  = (int)((blockIdx.x * blockDim.x + threadIdx.x) >> 5);
    const int lane = threadIdx.x & 31;
    if (wid >= N_EDGES) return;

    const int s = src[wid];
    const int d = dst[wid];
    const int r = etype[wid];

    const float4* mp = (const float4*)(xW + ((size_t)r * N_NODES + s) * D);
    float4 v = mp[lane];

    float* op = mid + (size_t)d * D + lane * 4;
    unsafeAtomicAdd(op + 0, v.x);
    unsafeAtomicAdd(op + 1, v.y);
    unsafeAtomicAdd(op + 2, v.z);
    unsafeAtomicAdd(op + 3, v.w);
}

// ---------------------------------------------------------------------------
// Kernel 3: fused MLP. Block = 8 waves, handles 32 rows (2 subtiles of 16).
// Phase 1: h = relu([x | mid] @ W1 + b1) -> LDS (row stride 258 floats).
// Phase 2: out = h @ W2 + b2 -> d_out.  mid aliases d_out (disjoint rows per
// block; __syncthreads separates the read and write phases).
// ---------------------------------------------------------------------------
#define HSTRIDE 258

__global__ __launch_bounds__(256) void k_mlp(
    const float* __restrict__ x, const float* mid,
    const float* __restrict__ W1, const float* __restrict__ b1,
    const float* __restrict__ W2, const float* __restrict__ b2,
    float* out)
{
    __shared__ float hl[32 * HSTRIDE];

    const int wave = threadIdx.x >> 5;
    const int lane = threadIdx.x & 31;
    const int m  = lane & 15;
    const int hi = lane >> 4;
    const int kh = hi * 2;
    const int t0 = blockIdx.x * 2;

    // ---- phase 1: h tiles (16 col-tiles of 16 over 2 subtiles = 32 jobs) ----
    #pragma unroll
    for (int j = 0; j < 4; ++j) {
        const int job = wave * 4 + j;
        const int s   = job >> 4;             // subtile 0/1
        const int ct  = job & 15;             // col tile 0..15 (256 cols)
        const int gt  = t0 + s;
        if (gt < N_TILES) {                   // wave-uniform
            const int r0 = gt * 16;
            v8f acc;
            {
                float bv = b1[ct * 16 + m];
                #pragma unroll
                for (int v = 0; v < 8; ++v) acc[v] = bv;
            }
            const float* ax = x   + (size_t)(r0 + m) * D;
            const float* am = mid + (size_t)(r0 + m) * D;
            for (int kc = 0; kc < 32; ++kc) {         // k = 0..127 from x
                const int kk = kc * 4 + kh;
                v2f a; a[0] = ax[kk]; a[1] = ax[kk + 1];
                v2f b;
                b[0] = W1[(size_t)kk * 256 + ct * 16 + m];
                b[1] = W1[(size_t)(kk + 1) * 256 + ct * 16 + m];
                acc = wmma4(a, b, acc);
            }
            for (int kc = 0; kc < 32; ++kc) {         // k = 128..255 from mid
                const int kk = kc * 4 + kh;
                const int kg = 128 + kk;
                v2f a; a[0] = am[kk]; a[1] = am[kk + 1];
                v2f b;
                b[0] = W1[(size_t)kg * 256 + ct * 16 + m];
                b[1] = W1[(size_t)(kg + 1) * 256 + ct * 16 + m];
                acc = wmma4(a, b, acc);
            }
            #pragma unroll
            for (int v = 0; v < 8; ++v) {
                const int lr = s * 16 + v + hi * 8;
                float val = acc[v];
                hl[lr * HSTRIDE + ct * 16 + m] = val > 0.0f ? val : 0.0f;
            }
        }
    }
    __syncthreads();

    // ---- phase 2: out tiles (8 col-tiles of 16 over 2 subtiles = 16 jobs) ----
    #pragma unroll
    for (int j = 0; j < 2; ++j) {
        const int job = wave * 2 + j;
        const int s   = job >> 3;
        const int ct  = job & 7;
        const int gt  = t0 + s;
        if (gt < N_TILES) {                   // wave-uniform
            const int r0 = gt * 16;
            v8f acc;
            {
                float bv = b2[ct * 16 + m];
                #pragma unroll
                for (int v = 0; v < 8; ++v) acc[v] = bv;
            }
            const float* hrow = hl + (s * 16 + m) * HSTRIDE;
            for (int kc = 0; kc < 64; ++kc) {         // k = 0..255 from LDS
                const int kk = kc * 4 + kh;
                v2f a = *(const v2f*)(hrow + kk);
                v2f b;
                b[0] = W2[(size_t)kk * D + ct * 16 + m];
                b[1] = W2[(size_t)(kk + 1) * D + ct * 16 + m];
                acc = wmma4(a, b, acc);
            }
            #pragma unroll
            for (int v = 0; v < 8; ++v)
                out[(size_t)(r0 + v + hi * 8) * D + ct * 16 + m] = acc[v];
        }
    }
}

// ---------------------------------------------------------------------------
extern "C" void kernel_launch(void* const* d_in, const int* in_sizes, int n_in,
                              void* d_out, int out_size, void* d_ws, size_t ws_size,
                              hipStream_t stream) {
    const float* x     = (const float*)d_in[0];
    const int*   src   = (const int*)  d_in[1];
    const int*   dst   = (const int*)  d_in[2];
    const int*   etype = (const int*)  d_in[3];
    const float* Wrel  = (const float*)d_in[4];
    const float* Wloop = (const float*)d_in[5];
    const float* brel  = (const float*)d_in[6];
    const float* W1    = (const float*)d_in[7];
    const float* b1    = (const float*)d_in[8];
    const float* W2    = (const float*)d_in[9];
    const float* b2    = (const float*)d_in[10];

    float* out = (float*)d_out;
    float* xW  = (float*)d_ws;   // 8 * 50000 * 128 floats = 204.8 MB scratch

    // 1) xW[r] = x @ Wrel[r]; mid(=d_out) = x @ Wloop + brel
    dim3 g1((N_TILES + 3) / 4, N_REL + 1);
    k_transform<<<g1, 128, 0, stream>>>(x, Wrel, Wloop, brel, xW, out);

    // 2) d_out += segment_sum over edges of xW[etype, src]
    k_scatter<<<(N_EDGES * 32) / 256, 256, 0, stream>>>(xW, src, dst, etype, out);

    // 3) d_out = relu([x | d_out] @ W1 + b1) @ W2 + b2
    k_mlp<<<(N_TILES + 1) / 2, 256, 0, stream>>>(x, out, W1, b1, W2, b2, out);
}